// WeightedFusion_76227079569440
// MI455X (gfx1250) — compile-verified
//
#include <hip/hip_runtime.h>
#include <math.h>

// Problem shape: predictions [S=4, B=8, C=9, H=512, W=512] f32, w [S,C], b [S]
#define S_ 4
#define B_ 8
#define C_ 9
#define HW_ (512 * 512)          // 262144, divisible by TILE_PIX
#define TILE_PIX 1024            // pixels per workgroup tile
#define THREADS 256              // 8 wave32; 4 pixels (float4) per thread
// LDS tile: [S][C][TILE_PIX] f32 = 4*9*1024*4 = 147456 B (fits 320KB/WGP)
#define LDS_BYTES (S_ * C_ * TILE_PIX * 4)

typedef __attribute__((ext_vector_type(4))) float        f32x4;
typedef __attribute__((ext_vector_type(4))) unsigned int u32x4;
typedef __attribute__((ext_vector_type(8))) int          i32x8;
typedef __attribute__((ext_vector_type(4))) int          i32x4;

__device__ __forceinline__ f32x4 vmax4(f32x4 a, f32x4 b) {
  f32x4 r;
  r.x = fmaxf(a.x, b.x); r.y = fmaxf(a.y, b.y);
  r.z = fmaxf(a.z, b.z); r.w = fmaxf(a.w, b.w);
  return r;
}
__device__ __forceinline__ f32x4 vexp4(f32x4 a) {
  f32x4 r;
  r.x = __expf(a.x); r.y = __expf(a.y);
  r.z = __expf(a.z); r.w = __expf(a.w);
  return r;
}

__global__ __launch_bounds__(THREADS) void WeightedFusion_tdm_kernel(
    const float* __restrict__ pred,   // [S,B,C,H,W]
    const float* __restrict__ wconv,  // [S,C]
    const float* __restrict__ bias,   // [S]
    float* __restrict__ out)          // [B,C,H,W]
{
  extern __shared__ float lds[];      // [S_][C_][TILE_PIX]

  const int tile = blockIdx.x;
  const int p0   = tile * TILE_PIX;   // flat pixel index over B*H*W
  const int b    = p0 / HW_;          // tiles never cross b (HW_ % TILE_PIX == 0)
  const int hw0  = p0 - b * HW_;

  // ---------------- TDM: one 3D tile DMA Global -> LDS per block -------------
  // x: 1024 contiguous pixels; y: c in [0,9) stride HW; z: s in [0,4) stride B*C*HW
  if (threadIdx.x < 32) {             // wave 0 issues; TDM is a wave-level op
    const unsigned long long ga =
        (unsigned long long)(const void*)pred +
        4ull * ((unsigned long long)b * (unsigned long long)(C_) * HW_ +
                (unsigned long long)hw0);
    const unsigned lds_base = (unsigned)(unsigned long long)(void*)&lds[0];

    // D# group 0: [1:0]=count=1 | [63:32]=lds_addr | [120:64]=global_addr | type=2
    u32x4 g0 = {
        1u,
        lds_base,
        (unsigned)(ga & 0xFFFFFFFFull),
        (unsigned)((ga >> 32) & 0x1FFFFFFull) | (2u << 30)
    };
    // D# group 1
    const long long d1stride = (long long)B_ * C_ * HW_;   // 18874368
    i32x8 g1 = {
        (int)(2u << 16),                        // data_size = 2 (4 bytes)
        (int)((TILE_PIX & 0xFFFF) << 16),       // tensor_dim0 lo16
        (int)(C_ << 16),                        // tensor_dim0 hi16(0) | tensor_dim1 lo16
        (int)(TILE_PIX << 16),                  // tensor_dim1 hi16(0) | tile_dim0
        (int)(C_ | (S_ << 16)),                 // tile_dim1 | tile_dim2
        (int)HW_,                               // tensor_dim0_stride lo32
        (int)((d1stride & 0xFFFFLL) << 16),     // d0stride hi16(0) | d1stride lo16
        (int)(d1stride >> 16)                   // d1stride hi32
    };
    i32x4 g2 = { S_, 0, 0, 0 };                 // tensor_dim2=4; dim3/tile_dim3 unused
    i32x4 g3 = { 0, 0, 0, 0 };
    i32x8 gx = { 0, 0, 0, 0, 0, 0, 0, 0 };      // extra arg of 6-arg builtin (zero)

    __builtin_amdgcn_tensor_load_to_lds(g0, g1, g2, g3, gx, /*cpol=*/0);
    __builtin_amdgcn_s_wait_tensorcnt(0);
  }
  __syncthreads();

  // ---------------- compute: softmax over S, weighted fuse -------------------
  const int x0 = threadIdx.x * 4;     // this thread's 4 pixels inside the tile

  // uniform-address weight/bias loads -> scalar loads
  float wl[S_][C_];
  float bl[S_];
#pragma unroll
  for (int s = 0; s < S_; ++s) {
    bl[s] = bias[s];
#pragma unroll
    for (int c = 0; c < C_; ++c) wl[s][c] = wconv[s * C_ + c];
  }

  // logits wm[s] = b[s] + sum_c w[s,c] * pred[s,c]  (per pixel component)
  f32x4 wm[S_];
#pragma unroll
  for (int s = 0; s < S_; ++s) {
    f32x4 acc = { bl[s], bl[s], bl[s], bl[s] };
#pragma unroll
    for (int c = 0; c < C_; ++c) {
      const f32x4 v = *(const f32x4*)&lds[(s * C_ + c) * TILE_PIX + x0];
      acc += wl[s][c] * v;
    }
    wm[s] = acc;
  }

  // stable softmax across S
  f32x4 m = wm[0];
#pragma unroll
  for (int s = 1; s < S_; ++s) m = vmax4(m, wm[s]);

  f32x4 wt[S_];
  f32x4 sum = { 0.f, 0.f, 0.f, 0.f };
#pragma unroll
  for (int s = 0; s < S_; ++s) { wt[s] = vexp4(wm[s] - m); sum += wt[s]; }
  f32x4 inv;
  inv.x = __frcp_rn(sum.x); inv.y = __frcp_rn(sum.y);
  inv.z = __frcp_rn(sum.z); inv.w = __frcp_rn(sum.w);
#pragma unroll
  for (int s = 0; s < S_; ++s) wt[s] *= inv;

  // fused[c] = sum_s wt[s] * pred[s,c]; stream out with NT stores
  const size_t obase = (size_t)b * C_ * HW_ + (size_t)hw0 + (size_t)x0;
#pragma unroll
  for (int c = 0; c < C_; ++c) {
    f32x4 acc = { 0.f, 0.f, 0.f, 0.f };
#pragma unroll
    for (int s = 0; s < S_; ++s)
      acc += wt[s] * (*(const f32x4*)&lds[(s * C_ + c) * TILE_PIX + x0]);
    __builtin_nontemporal_store(acc, (f32x4*)&out[obase + (size_t)c * HW_]);
  }
}

extern "C" void kernel_launch(void* const* d_in, const int* in_sizes, int n_in,
                              void* d_out, int out_size, void* d_ws, size_t ws_size,
                              hipStream_t stream) {
  const float* pred  = (const float*)d_in[0];  // [S,B,C,H,W]
  const float* wconv = (const float*)d_in[1];  // [S,C]
  const float* bias  = (const float*)d_in[2];  // [S]
  float* out = (float*)d_out;                  // [B,C,H,W]

  const int n_tiles = (B_ * HW_) / TILE_PIX;   // 2048 blocks
  hipLaunchKernelGGL(WeightedFusion_tdm_kernel,
                     dim3(n_tiles), dim3(THREADS), LDS_BYTES, stream,
                     pred, wconv, bias, out);
}